// SparseUnpool2d_20512763805963
// MI455X (gfx1250) — compile-verified
//
#include <hip/hip_runtime.h>
#include <cstdint>

// SparseUnpool2d for fixed shapes:
//   pooled : (16, 64, 128, 128) f32
//   sparse : (16, 64, 256, 256) f32
//   out    : (16, 64, 256, 256) f32
//   out[b,c,h,w] = (pooled[b,c,h/2,w/2] > 0.5) ? sparse[b,c,h,w] : 0
//
// Bandwidth-bound (576 MiB @ 23.3 TB/s ~ 25us). Strategy:
//  - block = 8 output rows x 256 cols of one (b,c) plane (4 pooled rows)
//  - stage the 4 pooled rows (2 KiB) into LDS via one per-lane
//    global_load_async_to_lds_b64 (ASYNCcnt path), wait, barrier
//  - stream sparse/out with non-temporal b128 loads/stores

typedef float v4f __attribute__((ext_vector_type(4)));
typedef float v2f __attribute__((ext_vector_type(2)));

#define OUT_H   256
#define OUT_W   256
#define POOL_H  128
#define POOL_W  128
#define TILE_H  8                      // output rows per block
#define TILE_PR (TILE_H / 2)           // pooled rows per block = 4
#define THREADS 256
#define W4      (OUT_W / 4)            // float4 per output row = 64
#define TILE_V4 (TILE_H * W4)          // float4 per tile = 512
#define HTILES  (OUT_H / TILE_H)       // 32 tiles per (b,c) plane

__global__ __launch_bounds__(THREADS)
void sparse_unpool2d_kernel(const float* __restrict__ pooled,
                            const float* __restrict__ sparse,
                            float* __restrict__ out)
{
    __shared__ float lds_pool[TILE_PR * POOL_W];   // 4 rows * 128 f32 = 2 KiB

    const int t     = threadIdx.x;
    const int blk   = blockIdx.x;
    const int htile = blk & (HTILES - 1);          // which 8-row band
    const int bc    = blk >> 5;                    // (b*C + c) plane index

    // ---- Stage 4 pooled rows into LDS with async global->LDS copy ----
    // 512 dwords total: each of the 256 lanes copies one b64 (8 bytes).
    {
        const float* gsrc =
            pooled + (long long)bc * (POOL_H * POOL_W)
                   + (long long)htile * (TILE_PR * POOL_W);
        unsigned goff = (unsigned)t * 8u;                          // byte offset (GVS vaddr)
        unsigned lofs = (unsigned)(uintptr_t)(&lds_pool[0]) + (unsigned)t * 8u; // LDS byte addr
        asm volatile("global_load_async_to_lds_b64 %0, %1, %2"
                     :
                     : "v"(lofs), "v"(goff), "s"(gsrc)
                     : "memory");
#if __has_builtin(__builtin_amdgcn_s_wait_asynccnt)
        __builtin_amdgcn_s_wait_asynccnt(0);
#else
        asm volatile("s_wait_asynccnt 0" ::: "memory");
#endif
    }
    __syncthreads();

    // ---- Stream 8 rows x 256 cols as 512 float4 (2 per thread) ----
    const long long outBase =
        (long long)bc * (OUT_H * OUT_W) + (long long)htile * (TILE_H * OUT_W);
    const v4f* __restrict__ sp4 = reinterpret_cast<const v4f*>(sparse + outBase);
    v4f* __restrict__ ot4       = reinterpret_cast<v4f*>(out + outBase);

#pragma unroll
    for (int k = 0; k < TILE_V4 / THREADS; ++k) {
        const int idx  = k * THREADS + t;      // 0..511
        const int w4   = idx & (W4 - 1);       // float4 column (w = 4*w4)
        const int hrow = idx >> 6;             // 0..7 within tile
        const int prow = hrow >> 1;            // pooled row within tile

        // two pooled values cover this float4: pooled w = 2*w4, 2*w4+1
        const v2f p = *reinterpret_cast<const v2f*>(&lds_pool[prow * POOL_W + 2 * w4]);

        const v4f s = __builtin_nontemporal_load(&sp4[idx]);

        v4f r;
        r.x = (p.x > 0.5f) ? s.x : 0.0f;
        r.y = (p.x > 0.5f) ? s.y : 0.0f;
        r.z = (p.y > 0.5f) ? s.z : 0.0f;
        r.w = (p.y > 0.5f) ? s.w : 0.0f;

        __builtin_nontemporal_store(r, &ot4[idx]);
    }
}

extern "C" void kernel_launch(void* const* d_in, const int* in_sizes, int n_in,
                              void* d_out, int out_size, void* d_ws, size_t ws_size,
                              hipStream_t stream)
{
    const float* pooled = (const float*)d_in[0];   // (16,64,128,128)
    const float* sparse = (const float*)d_in[1];   // (16,64,256,256)
    float* out          = (float*)d_out;           // (16,64,256,256)

    // out_size = 16*64*256*256 = 67,108,864 ; each block covers 8*256 = 2048 floats
    const int blocks = out_size / (TILE_H * OUT_W); // 32768

    sparse_unpool2d_kernel<<<blocks, THREADS, 0, stream>>>(pooled, sparse, out);
}